// MFA_block_40999757807684
// MI455X (gfx1250) — compile-verified
//
#include <hip/hip_runtime.h>
#include <hip/hip_bf16.h>

typedef __attribute__((ext_vector_type(2))) float v2f;
typedef __attribute__((ext_vector_type(8))) float v8f;
typedef int v4i __attribute__((vector_size(4 * sizeof(int))));   // b128 payload

#define BNROWS 8192
#define DHI 512
#define DLO 256
#define BN_EPS 1e-5f

// --- CDNA5 async global->LDS path (guarded; falls back to reg-staged copy) --
#if defined(__HIP_DEVICE_COMPILE__) && __has_builtin(__builtin_amdgcn_global_load_async_to_lds_b128)
#define HAVE_ASYNC_LDS 1
#else
#define HAVE_ASYNC_LDS 0
#endif

#define GLB_AS1(p) ((__attribute__((address_space(1))) v4i*)(p))
#define LDS_AS3(p) ((__attribute__((address_space(3))) v4i*)(p))

__device__ __forceinline__ void wait_async_zero() {
#if HAVE_ASYNC_LDS
#if __has_builtin(__builtin_amdgcn_s_wait_asynccnt)
    __builtin_amdgcn_s_wait_asynccnt(0);
#else
    asm volatile("s_wait_asynccnt 0" ::: "memory");
#endif
#endif
}

// ---------------------------------------------------------------------------
// C(M,N) = A(M,K) @ B(K,N) + bias   (all row-major f32, bias may be nullptr)
// block = 256 threads (8 wave32). Wave w computes rows [(by*8+w)*16, +16),
// cols [bx*64, +64) as four 16x16 f32 WMMA tiles. The shared 16x64 B chunk is
// double-buffered in LDS via async global->LDS b128 copies that overlap the
// WMMA chain. K % 16 == 0, N % 64 == 0, M % 128 == 0.
// ---------------------------------------------------------------------------
__global__ __launch_bounds__(256) void gemm_f32_wmma(
    const float* __restrict__ A, const float* __restrict__ B,
    const float* __restrict__ bias, float* __restrict__ C,
    int M, int N, int K)
{
    __shared__ float Bs[2][16 * 64];           // 8 KB, double-buffered B chunk

    const int t    = threadIdx.x;
    const int lane = t & 31;
    const int wave = t >> 5;
    const int m0   = ((int)blockIdx.y * 8 + wave) * 16;
    const int n0   = (int)blockIdx.x * 64;
    const int ml   = lane & 15;
    const int half = lane >> 4;

    // async B staging: thread t copies row (t>>4), cols 4*(t&15)..+4
    const int lr = t >> 4;                     // 0..15 row within K chunk
    const int lc = (t & 15) * 4;               // 0..60 col within 64-wide tile
    const float* gB = B + (size_t)lr * N + n0 + lc;
    float* sB = &Bs[0][lr * 64 + lc];

    v8f acc0 = {}, acc1 = {}, acc2 = {}, acc3 = {};
    const float* Arow = A + (size_t)(m0 + ml) * K + 2 * half;

    // preload chunk 0 into buffer 0
#if HAVE_ASYNC_LDS
    __builtin_amdgcn_global_load_async_to_lds_b128(GLB_AS1(gB), LDS_AS3(sB), 0, 0);
#else
    *(float4*)sB = *(const float4*)gB;
#endif
    wait_async_zero();
    __syncthreads();

    int buf = 0;
    for (int kc = 0; kc < K; kc += 16) {
        // kick off the next chunk's DMA into the other buffer
        const bool more = (kc + 16) < K;
        if (more) {
            const float* gnext = gB + (size_t)(kc + 16) * N;
            float* snext = sB + (buf ? -1024 : 1024);
#if HAVE_ASYNC_LDS
            __builtin_amdgcn_global_load_async_to_lds_b128(GLB_AS1(gnext), LDS_AS3(snext), 0, 0);
#else
            *(float4*)snext = *(const float4*)gnext;
#endif
        }

        const float* __restrict__ bs = &Bs[buf][ml];
#pragma unroll
        for (int kk = 0; kk < 16; kk += 4) {
            v2f a = *(const v2f*)(Arow + kc + kk);
            __builtin_prefetch(Arow + kc + kk + 32, 0, 3);   // near-scope prefetch
            const float* bp = bs + (kk + 2 * half) * 64;
            v2f b0, b1, b2, b3;
            b0.x = bp[0];  b0.y = bp[64 + 0];
            b1.x = bp[16]; b1.y = bp[64 + 16];
            b2.x = bp[32]; b2.y = bp[64 + 32];
            b3.x = bp[48]; b3.y = bp[64 + 48];
            acc0 = __builtin_amdgcn_wmma_f32_16x16x4_f32(false, a, false, b0, (short)0, acc0, false, false);
            acc1 = __builtin_amdgcn_wmma_f32_16x16x4_f32(false, a, false, b1, (short)0, acc1, false, false);
            acc2 = __builtin_amdgcn_wmma_f32_16x16x4_f32(false, a, false, b2, (short)0, acc2, false, false);
            acc3 = __builtin_amdgcn_wmma_f32_16x16x4_f32(false, a, false, b3, (short)0, acc3, false, false);
        }

        if (more) wait_async_zero();   // DMA for next chunk has landed
        __syncthreads();               // everyone done reading current buffer
        buf ^= 1;
    }

    // C layout: VGPR r -> row m0 + r + 8*half, col n0 + nt*16 + ml
    const int colb = n0 + ml;
    float bv0 = bias ? bias[colb +  0] : 0.0f;
    float bv1 = bias ? bias[colb + 16] : 0.0f;
    float bv2 = bias ? bias[colb + 32] : 0.0f;
    float bv3 = bias ? bias[colb + 48] : 0.0f;
#pragma unroll
    for (int r = 0; r < 8; ++r) {
        const size_t row = (size_t)(m0 + r + 8 * half) * N;
        C[row + colb +  0] = acc0[r] + bv0;
        C[row + colb + 16] = acc1[r] + bv1;
        C[row + colb + 32] = acc2[r] + bv2;
        C[row + colb + 48] = acc3[r] + bv3;
    }
}

// ---------------------------------------------------------------------------
// C(M,N) = scale * (Aᵀ @ B)   where A is stored (K,M) row-major (so we need
// A[m][k] = Astore[k][m]), B is (K,N) row-major. Used for  M = φᵀ g / 8192.
// ---------------------------------------------------------------------------
__global__ __launch_bounds__(256) void gemm_at_f32_wmma(
    const float* __restrict__ Astore, const float* __restrict__ B,
    float* __restrict__ C, int M, int N, int K, float scale)
{
    const int lane = threadIdx.x & 31;
    const int wave = threadIdx.x >> 5;
    const int m0   = ((int)blockIdx.y * 8 + wave) * 16;
    const int n0   = (int)blockIdx.x * 64;
    const int ml   = lane & 15;
    const int half = lane >> 4;

    v8f acc0 = {}, acc1 = {}, acc2 = {}, acc3 = {};

    const float* Acol = Astore + (size_t)(2 * half) * M + (m0 + ml); // A[k][m]
    const float* Brow = B + (size_t)(2 * half) * N + n0 + ml;

    for (int k = 0; k < K; k += 4) {
        const float* ap = Acol + (size_t)k * M;
        v2f a; a.x = ap[0]; a.y = ap[M];
        const float* bp = Brow + (size_t)k * N;
        v2f b0, b1, b2, b3;
        b0.x = bp[0];  b0.y = bp[N + 0];
        b1.x = bp[16]; b1.y = bp[N + 16];
        b2.x = bp[32]; b2.y = bp[N + 32];
        b3.x = bp[48]; b3.y = bp[N + 48];
        acc0 = __builtin_amdgcn_wmma_f32_16x16x4_f32(false, a, false, b0, (short)0, acc0, false, false);
        acc1 = __builtin_amdgcn_wmma_f32_16x16x4_f32(false, a, false, b1, (short)0, acc1, false, false);
        acc2 = __builtin_amdgcn_wmma_f32_16x16x4_f32(false, a, false, b2, (short)0, acc2, false, false);
        acc3 = __builtin_amdgcn_wmma_f32_16x16x4_f32(false, a, false, b3, (short)0, acc3, false, false);
    }

    const int colb = n0 + ml;
#pragma unroll
    for (int r = 0; r < 8; ++r) {
        const size_t row = (size_t)(m0 + r + 8 * half) * N;
        C[row + colb +  0] = acc0[r] * scale;
        C[row + colb + 16] = acc1[r] * scale;
        C[row + colb + 32] = acc2[r] * scale;
        C[row + colb + 48] = acc3[r] * scale;
    }
}

// ---------------------------------------------------------------------------
// BatchNorm statistics over the row dimension: one block per feature column.
// ---------------------------------------------------------------------------
__global__ __launch_bounds__(256) void bn_stats(
    const float* __restrict__ wy, float* __restrict__ mean,
    float* __restrict__ rsig)
{
    __shared__ float s1[256];
    __shared__ float s2[256];
    const int j = blockIdx.x;            // feature 0..DHI-1
    const int t = threadIdx.x;
    float a = 0.0f, b = 0.0f;
    for (int i = t; i < BNROWS; i += 256) {
        float v = wy[(size_t)i * DHI + j];
        a += v;
        b += v * v;
    }
    s1[t] = a; s2[t] = b;
    __syncthreads();
    for (int s = 128; s > 0; s >>= 1) {
        if (t < s) { s1[t] += s1[t + s]; s2[t] += s2[t + s]; }
        __syncthreads();
    }
    if (t == 0) {
        float m   = s1[0] * (1.0f / BNROWS);
        float var = s2[0] * (1.0f / BNROWS) - m * m;
        mean[j] = m;
        rsig[j] = rsqrtf(var + BN_EPS);
    }
}

// ---------------------------------------------------------------------------
// z = (w_y - mean) * rsig * gamma + beta + x_h
// ---------------------------------------------------------------------------
__global__ __launch_bounds__(256) void bn_apply_residual(
    const float* __restrict__ wy, const float* __restrict__ xh,
    const float* __restrict__ mean, const float* __restrict__ rsig,
    const float* __restrict__ gamma, const float* __restrict__ beta,
    float* __restrict__ out)
{
    const size_t idx = (size_t)blockIdx.x * 256 + threadIdx.x;
    const int j = (int)(idx & (DHI - 1));
    out[idx] = (wy[idx] - mean[j]) * rsig[j] * gamma[j] + beta[j] + xh[idx];
}

// ---------------------------------------------------------------------------
extern "C" void kernel_launch(void* const* d_in, const int* in_sizes, int n_in,
                              void* d_out, int out_size, void* d_ws, size_t ws_size,
                              hipStream_t stream)
{
    const float* x_h  = (const float*)d_in[0];
    const float* x_l  = (const float*)d_in[1];
    const float* g_w  = (const float*)d_in[2];
    const float* g_b  = (const float*)d_in[3];
    const float* th_w = (const float*)d_in[4];
    const float* th_b = (const float*)d_in[5];
    const float* ph_w = (const float*)d_in[6];
    const float* ph_b = (const float*)d_in[7];
    const float* w_w  = (const float*)d_in[8];
    const float* w_b  = (const float*)d_in[9];
    const float* gam  = (const float*)d_in[10];
    const float* bet  = (const float*)d_in[11];
    float* out = (float*)d_out;
    float* ws  = (float*)d_ws;

    // workspace layout (floats); w_y reuses the dead g/phi region
    float* theta = ws;                 // 8192*256 = 2,097,152
    float* gx    = ws + 2097152;       // 2,097,152
    float* phix  = ws + 4194304;       // 2,097,152
    float* y     = ws + 6291456;       // 2,097,152
    float* Mm    = ws + 8388608;       // 256*256 = 65,536
    float* mean  = ws + 8454144;       // 512
    float* rsig  = ws + 8454656;       // 512
    float* wy    = ws + 2097152;       // 8192*512 = 4,194,304 (overlaps gx+phix)

    const dim3 blk(256);

    // theta = x_h @ th_w + th_b          (8192 x 256, K=512)
    gemm_f32_wmma<<<dim3(256 / 64, 8192 / 128), blk, 0, stream>>>(
        x_h, th_w, th_b, theta, 8192, 256, 512);
    // g = x_l @ g_w + g_b                (8192 x 256, K=256)
    gemm_f32_wmma<<<dim3(256 / 64, 8192 / 128), blk, 0, stream>>>(
        x_l, g_w, g_b, gx, 8192, 256, 256);
    // phi = x_l @ ph_w + ph_b            (8192 x 256, K=256)
    gemm_f32_wmma<<<dim3(256 / 64, 8192 / 128), blk, 0, stream>>>(
        x_l, ph_w, ph_b, phix, 8192, 256, 256);
    // Mm = (phiᵀ @ g) / 8192             (256 x 256, K=8192)
    gemm_at_f32_wmma<<<dim3(256 / 64, 256 / 128), blk, 0, stream>>>(
        phix, gx, Mm, 256, 256, 8192, 1.0f / 8192.0f);
    // y = theta @ Mm                     (8192 x 256, K=256)   [no bias]
    gemm_f32_wmma<<<dim3(256 / 64, 8192 / 128), blk, 0, stream>>>(
        theta, Mm, nullptr, y, 8192, 256, 256);
    // w_y = y @ w_w + w_b                (8192 x 512, K=256)
    gemm_f32_wmma<<<dim3(512 / 64, 8192 / 128), blk, 0, stream>>>(
        y, w_w, w_b, wy, 8192, 512, 256);
    // BatchNorm stats + apply + residual
    bn_stats<<<DHI, blk, 0, stream>>>(wy, mean, rsig);
    bn_apply_residual<<<(BNROWS * DHI) / 256, blk, 0, stream>>>(
        wy, x_h, mean, rsig, gam, bet, out);
}